// FuzzyMLP_82231443849327
// MI455X (gfx1250) — compile-verified
//
#include <hip/hip_runtime.h>

// ---------------------------------------------------------------------------
// FuzzyMLP (interval-product min/max reduction), MI455X / gfx1250.
//
// Exact decomposition (wm=(wl+wr)/2, wd=(wr-wl)/2>=0; wl=w_b-relu(w_a),
// wr=w_b+relu(w_c)):
//   min over 4 corners = xm*wm - xa*wd - |hdx*wm - hda*wd|
//   max over 4 corners = xm*wm + xa*wd + |hdx*wm + hda*wd|
// with xm=(hl+hr)/2, xa=(|hl|+|hr|)/2, hdx=(hl-hr)/2, hda=(|hl|-|hr|)/2.
//
// => out = (two f32 GEMMs on the WMMA pipe) +/- (5-VALU-op/triple residual).
// GEMM part: V_WMMA_F32_16X16X4_F32, one wave per 16x16 tile.
// Residual part: VALU-bound (67M triples * 5 ops), LDS-resident tiles staged
// with gfx1250 async global->LDS DMA (ASYNCcnt), 2n x 4o register tile.
// ---------------------------------------------------------------------------

typedef float v2f __attribute__((ext_vector_type(2)));
typedef float v4  __attribute__((ext_vector_type(4)));
typedef float v8f __attribute__((ext_vector_type(8)));

// Async builtin pointer params are GCC-vector typed (per hipcc diagnostic).
typedef int vi4 __attribute__((__vector_size__(16)));
typedef __attribute__((address_space(1))) vi4 as1_vi4;
typedef __attribute__((address_space(3))) vi4 as3_vi4;

#define NROWS 1024
#define IND   256
#define OUTD  256

#define TN 16          // n-rows per residual block
#define TO 64          // o-cols per residual block
#define NTHREADS 128   // 4 waves (wave32); thread tile = 2n x 4o

#if defined(__has_builtin)
#if __has_builtin(__builtin_amdgcn_global_load_async_to_lds_b128)
#define HAVE_ASYNC_LDS 1
#endif
#endif

__device__ __forceinline__ void cp128(const float* __restrict__ g, float* l) {
#ifdef HAVE_ASYNC_LDS
  __builtin_amdgcn_global_load_async_to_lds_b128(
      (as1_vi4*)const_cast<float*>(g), (as3_vi4*)l, /*offset=*/0, /*cpol=*/0);
#else
  *(v4*)l = *(const v4*)g;
#endif
}

__device__ __forceinline__ void stage_wait() {
#ifdef HAVE_ASYNC_LDS
#if defined(__has_builtin) && __has_builtin(__builtin_amdgcn_s_wait_asynccnt)
  __builtin_amdgcn_s_wait_asynccnt(0);
#else
  asm volatile("s_wait_asynccnt 0" ::: "memory");
#endif
#endif
  __syncthreads();
}

// --- prologue: midpoint/radius weight matrices -----------------------------
__global__ void prep_weights(const float* __restrict__ wb,
                             const float* __restrict__ wa,
                             const float* __restrict__ wc,
                             float* __restrict__ wm,
                             float* __restrict__ wd) {
  int idx = blockIdx.x * blockDim.x + threadIdx.x;
  if (idx < IND * OUTD) {
    float ra = fmaxf(wa[idx], 0.0f);
    float rc = fmaxf(wc[idx], 0.0f);
    wm[idx] = wb[idx] + 0.5f * (rc - ra);   // (wl+wr)/2
    wd[idx] = 0.5f * (rc + ra);             // (wr-wl)/2 >= 0
  }
}

// --- GEMM half on the WMMA pipe: base = xm@wm -/+ xa@wd + bias -------------
// One wave (32 lanes) per 16x16 output tile; K-loop of 64 x 2 WMMAs.
// A layout (f32 16x16x4): lane L holds row M=L%16, K = {0,1}+2*(L>=16).
// B layout: lane L holds col N=L%16, K = {0,1}+2*(L>=16).
// C/D layout: VGPR r <-> (M = r + 8*(L>=16), N = L%16).
__global__ void __launch_bounds__(32)
wmma_base(const float* __restrict__ hl, const float* __restrict__ hr,
          const float* __restrict__ wm, const float* __restrict__ wd,
          const float* __restrict__ bb, const float* __restrict__ ba,
          const float* __restrict__ bc, float* __restrict__ out) {
  const int L  = threadIdx.x;
  const int m0 = blockIdx.x * 16;
  const int n0 = blockIdx.y * 16;

  const int mrow  = m0 + (L & 15);
  const int ncol  = n0 + (L & 15);
  const int khalf = (L >> 4) * 2;          // 0 or 2

  v8f s1 = {};  // xm @ wm
  v8f s2 = {};  // xa @ wd
  for (int k = 0; k < IND; k += 4) {
    const int ka = k + khalf;
    v2f xl = *(const v2f*)&hl[mrow * IND + ka];
    v2f xr = *(const v2f*)&hr[mrow * IND + ka];
    v2f am, aa, bm, bd;
    am[0] = 0.5f * (xl[0] + xr[0]);
    am[1] = 0.5f * (xl[1] + xr[1]);
    aa[0] = 0.5f * (fabsf(xl[0]) + fabsf(xr[0]));
    aa[1] = 0.5f * (fabsf(xl[1]) + fabsf(xr[1]));
    bm[0] = wm[ka * OUTD + ncol];
    bm[1] = wm[(ka + 1) * OUTD + ncol];
    bd[0] = wd[ka * OUTD + ncol];
    bd[1] = wd[(ka + 1) * OUTD + ncol];
    s1 = __builtin_amdgcn_wmma_f32_16x16x4_f32(false, am, false, bm,
                                               (short)0, s1, false, false);
    s2 = __builtin_amdgcn_wmma_f32_16x16x4_f32(false, aa, false, bd,
                                               (short)0, s2, false, false);
  }

  const float bbv = bb[ncol];
  const float bl  = bbv - fmaxf(ba[ncol], 0.0f);
  const float br  = bbv + fmaxf(bc[ncol], 0.0f);
  const int mbase = m0 + ((L >> 4) << 3);
#pragma unroll
  for (int r = 0; r < 8; ++r) {
    const int g = mbase + r;
    out[g * OUTD + ncol]                = s1[r] - s2[r] + bl;
    out[NROWS * OUTD + g * OUTD + ncol] = s1[r] + s2[r] + br;
  }
}

// --- residual half on the VALU pipe ----------------------------------------
// out_l -= 0.5*sum_i |dx*wm - da*wd| ;  out_r += 0.5*sum_i |dx*wm + da*wd|
__global__ void __launch_bounds__(NTHREADS)
fuzzy_residual(const float* __restrict__ hl, const float* __restrict__ hr,
               const float* __restrict__ wmg, const float* __restrict__ wdg,
               float* __restrict__ out) {
  // LDS: wm/wd tiles [IND][TO] + raw hl/hr tiles [TN][IND] = 160 KB.
  extern __shared__ float smem[];
  float* LWm = smem;
  float* LWd = smem + IND * TO;
  float* LXl = smem + 2 * IND * TO;
  float* LXr = LXl + TN * IND;

  const int tid = threadIdx.x;
  const int o0  = blockIdx.x * TO;
  const int n0  = blockIdx.y * TN;

#pragma unroll 4
  for (int k = 0; k < (IND * TO / 4) / NTHREADS; ++k) {
    int lin = k * NTHREADS + tid;
    int i   = lin >> 4;
    int oq  = lin & 15;
    cp128(&wmg[i * OUTD + o0 + oq * 4], &LWm[i * TO + oq * 4]);
    cp128(&wdg[i * OUTD + o0 + oq * 4], &LWd[i * TO + oq * 4]);
  }
#pragma unroll
  for (int k = 0; k < (TN * IND / 4) / NTHREADS; ++k) {
    int lin = k * NTHREADS + tid;
    int n   = lin >> 6;
    int iq  = lin & 63;
    cp128(&hl[(n0 + n) * IND + iq * 4], &LXl[n * IND + iq * 4]);
    cp128(&hr[(n0 + n) * IND + iq * 4], &LXr[n * IND + iq * 4]);
  }
  stage_wait();

  const int oq = tid & 15;
  const int co = oq * 4;
  const int rn = (tid >> 4) * 2;

  float rm[2][4] = {};
  float rp[2][4] = {};

  for (int i = 0; i < IND; i += 4) {
    v4 xl0 = *(const v4*)&LXl[rn * IND + i];
    v4 xl1 = *(const v4*)&LXl[(rn + 1) * IND + i];
    v4 xr0 = *(const v4*)&LXr[rn * IND + i];
    v4 xr1 = *(const v4*)&LXr[(rn + 1) * IND + i];
#pragma unroll
    for (int u = 0; u < 4; ++u) {
      v4 wm4 = *(const v4*)&LWm[(i + u) * TO + co];
      v4 wd4 = *(const v4*)&LWd[(i + u) * TO + co];
      float dx0 = xl0[u] - xr0[u];
      float da0 = fabsf(xl0[u]) - fabsf(xr0[u]);
      float dx1 = xl1[u] - xr1[u];
      float da1 = fabsf(xl1[u]) - fabsf(xr1[u]);
#pragma unroll
      for (int c = 0; c < 4; ++c) {
        float t0 = da0 * wd4[c];
        rm[0][c] += fabsf(fmaf(dx0, wm4[c], -t0));
        rp[0][c] += fabsf(fmaf(dx0, wm4[c],  t0));
        float t1 = da1 * wd4[c];
        rm[1][c] += fabsf(fmaf(dx1, wm4[c], -t1));
        rp[1][c] += fabsf(fmaf(dx1, wm4[c],  t1));
      }
    }
  }

  // Read-modify-write the GEMM base written by wmma_base.
#pragma unroll
  for (int n = 0; n < 2; ++n) {
    const int g = n0 + rn + n;
    v4 ol  = *(v4*)&out[g * OUTD + o0 + co];
    v4 orr = *(v4*)&out[NROWS * OUTD + g * OUTD + o0 + co];
#pragma unroll
    for (int c = 0; c < 4; ++c) {
      ol[c]  = fmaf(-0.5f, rm[n][c], ol[c]);
      orr[c] = fmaf( 0.5f, rp[n][c], orr[c]);
    }
    *(v4*)&out[g * OUTD + o0 + co]                = ol;
    *(v4*)&out[NROWS * OUTD + g * OUTD + o0 + co] = orr;
  }
}

extern "C" void kernel_launch(void* const* d_in, const int* in_sizes, int n_in,
                              void* d_out, int out_size, void* d_ws, size_t ws_size,
                              hipStream_t stream) {
  (void)in_sizes; (void)n_in; (void)out_size; (void)ws_size;
  const float* hl = (const float*)d_in[0];
  const float* hr = (const float*)d_in[1];
  const float* wb = (const float*)d_in[2];
  const float* wa = (const float*)d_in[3];
  const float* wc = (const float*)d_in[4];
  const float* bb = (const float*)d_in[5];
  const float* ba = (const float*)d_in[6];
  const float* bc = (const float*)d_in[7];

  float* wm = (float*)d_ws;             // 256 KB
  float* wd = wm + IND * OUTD;          // 256 KB  (needs ws_size >= 512 KB)

  prep_weights<<<(IND * OUTD + 255) / 256, 256, 0, stream>>>(wb, wa, wc, wm, wd);

  dim3 ggrid(NROWS / 16, OUTD / 16);    // 64 x 16 = 1024 single-wave blocks
  wmma_base<<<ggrid, 32, 0, stream>>>(hl, hr, wm, wd, bb, ba, bc, (float*)d_out);

  dim3 rgrid(OUTD / TO, NROWS / TN);    // 4 x 64 = 256 blocks
  size_t smem_bytes = (size_t)(2 * IND * TO + 2 * TN * IND) * sizeof(float); // 160 KB
  fuzzy_residual<<<rgrid, NTHREADS, smem_bytes, stream>>>(hl, hr, wm, wd,
                                                          (float*)d_out);
}